// HungarianMatcher_661424963855
// MI455X (gfx1250) — compile-verified
//
#include <hip/hip_runtime.h>

#define BS 32
#define NQ 512
#define NC 16
#define NT 64
#define PTOT (BS * NQ)   // 16384 query rows
#define TTOT (BS * NT)   // 2048 target cols

typedef __attribute__((ext_vector_type(2))) float v2f;
typedef __attribute__((ext_vector_type(8))) float v8f;

// ---------------------------------------------------------------------------
// Kernel 1: row-wise softmax over NC=16 classes, one thread per row.
// ---------------------------------------------------------------------------
__global__ void hm_softmax16(const float* __restrict__ logits,
                             float* __restrict__ probs) {
  int row = blockIdx.x * blockDim.x + threadIdx.x;
  if (row >= PTOT) return;
  const float* lr = logits + row * NC;
  float m = lr[0];
#pragma unroll
  for (int c = 1; c < NC; ++c) m = fmaxf(m, lr[c]);
  float e[NC];
  float s = 0.f;
#pragma unroll
  for (int c = 0; c < NC; ++c) {
    e[c] = __expf(lr[c] - m);
    s += e[c];
  }
  float inv = 1.f / s;
  float* pr = probs + row * NC;
#pragma unroll
  for (int c = 0; c < NC; ++c) pr[c] = e[c] * inv;
}

// ---------------------------------------------------------------------------
// Kernel 2: cost matrix. One wave per 16x16 tile of C[16384 x 2048].
// Class cost = probs @ onehot(tgt_ids)  via V_WMMA_F32_16X16X4_F32 (exact
// fp32 gather-as-GEMM), L1 terms fused in the epilogue from the accumulator
// fragment layout (VGPR r: lanes 0-15 -> M=r, lanes 16-31 -> M=8+r, N=lane%16).
// ---------------------------------------------------------------------------
__global__ __launch_bounds__(256) void hm_cost(
    const float* __restrict__ probs, const float* __restrict__ pred_boxes,
    const float* __restrict__ pred_angles, const int* __restrict__ tgt_ids,
    const float* __restrict__ tgt_bbox, const float* __restrict__ tgt_angles,
    float* __restrict__ out) {
  const int lane = threadIdx.x & 31;
  const int wave = threadIdx.x >> 5;
  const int tile = blockIdx.x * 8 + wave;
  const int numColTiles = TTOT / 16;  // 128
  const int rowTile = tile / numColTiles;
  const int colTile = tile - rowTile * numColTiles;
  const int p0 = rowTile * 16;
  const int n0 = colTile * 16;

  const int nl = n0 + (lane & 15);  // this lane's target column
  const int tid_n = tgt_ids[nl];
  const int m = lane & 15;
  const int hi = (lane >= 16) ? 1 : 0;

  v8f acc = {0.f, 0.f, 0.f, 0.f, 0.f, 0.f, 0.f, 0.f};
#pragma unroll
  for (int s = 0; s < 4; ++s) {
    const int k0 = 4 * s + 2 * hi;
    v2f a, b;
    // A fragment: 16x4 f32 slice of probs (rows p0..p0+15, K = k0,k0+1)
    a.x = probs[(p0 + m) * NC + k0];
    a.y = probs[(p0 + m) * NC + k0 + 1];
    // B fragment: 4x16 one-hot slice: B[k][n] = (tgt_ids[n] == k)
    b.x = (tid_n == k0) ? 1.0f : 0.0f;
    b.y = (tid_n == k0 + 1) ? 1.0f : 0.0f;
    acc = __builtin_amdgcn_wmma_f32_16x16x4_f32(false, a, false, b, (short)0,
                                                acc, false, false);
  }

  const float4 tb = ((const float4*)tgt_bbox)[nl];
  const float ta = tgt_angles[nl];

#pragma unroll
  for (int r = 0; r < 8; ++r) {
    const int pp = p0 + r + 8 * hi;  // query row for accumulator VGPR r
    const float4 pb = ((const float4*)pred_boxes)[pp];
    const float pa = pred_angles[pp];
    const float l1 = fabsf(pb.x - tb.x) + fabsf(pb.y - tb.y)   // centre
                   + fabsf(pb.z - tb.z) + fabsf(pb.w - tb.w)   // dims
                   + fabsf(pa - ta);                           // angle
    out[(size_t)pp * TTOT + nl] = l1 - 3.0f * acc[r];
  }
}

// ---------------------------------------------------------------------------
// Kernel 3: Hungarian (shortest augmenting path) on the transposed 64x512
// per-batch block. One wave per batch; columns strided across 32 lanes;
// min/argmin reduced with __shfl_xor. Cost block staged in dynamic LDS.
// ---------------------------------------------------------------------------
#define HM_INF 1e30f

__global__ void hm_lsa(const float* __restrict__ Cg,
                       float* __restrict__ row_out,
                       float* __restrict__ col_out) {
  const int b = blockIdx.x;
  const int lane = threadIdx.x;

  extern __shared__ char smem[];
  float* costT = (float*)smem;       // [64][512] transposed cost block
  float* v = costT + NT * NQ;        // [513]
  float* minv = v + (NQ + 1);        // [513]
  float* u = minv + (NQ + 1);        // [65]
  int* way = (int*)(u + (NT + 1));   // [513]
  int* p = way + (NQ + 1);           // [513]
  int* used = p + (NQ + 1);          // [513]

  // Stage transposed block: costT[t][q] = C[b, q, b*NT + t]
  for (int t = 0; t < NT; ++t)
    for (int q = lane; q < NQ; q += 32)
      costT[t * NQ + q] = Cg[(size_t)(b * NQ + q) * TTOT + b * NT + t];

  for (int j = lane; j <= NQ; j += 32) {
    v[j] = 0.f;
    p[j] = 0;
  }
  for (int j = lane; j <= NT; j += 32) u[j] = 0.f;
  __syncthreads();

  for (int i = 1; i <= NT; ++i) {
    if (lane == 0) p[0] = i;
    for (int j = lane; j <= NQ; j += 32) {
      minv[j] = HM_INF;
      used[j] = 0;
    }
    __syncthreads();
    int j0 = 0;
    while (true) {
      if (lane == 0) used[j0] = 1;
      __syncthreads();
      const int i0 = p[j0];
      const float ui0 = u[i0];
      float lmin = HM_INF;
      int lj = 0x7fffffff;
      for (int j = lane + 1; j <= NQ; j += 32) {
        if (!used[j]) {
          float cur = costT[(i0 - 1) * NQ + (j - 1)] - ui0 - v[j];
          if (cur < minv[j]) {
            minv[j] = cur;
            way[j] = j0;
          }
          float mv = minv[j];
          if (mv < lmin || (mv == lmin && j < lj)) {
            lmin = mv;
            lj = j;
          }
        }
      }
#pragma unroll
      for (int off = 16; off >= 1; off >>= 1) {
        float om = __shfl_xor(lmin, off, 32);
        int oj = __shfl_xor(lj, off, 32);
        if (om < lmin || (om == lmin && oj < lj)) {
          lmin = om;
          lj = oj;
        }
      }
      const float delta = lmin;
      const int j1 = lj;
      __syncthreads();
      for (int j = lane; j <= NQ; j += 32) {
        if (used[j]) {
          u[p[j]] += delta;  // matched rows are distinct -> no race
          v[j] -= delta;
        } else {
          minv[j] -= delta;
        }
      }
      __syncthreads();
      j0 = j1;
      if (p[j0] == 0) break;
    }
    // augment along alternating path (serial, tiny)
    if (lane == 0) {
      int jj = j0;
      while (jj != 0) {
        int jp = way[jj];
        p[jj] = p[jp];
        jj = jp;
      }
    }
    __syncthreads();
  }

  // Emit (query, target) pairs; j ascending == sorted by query index,
  // matching the reference's post-transpose ordering.
  if (lane == 0) {
    int k = 0;
    for (int j = 1; j <= NQ; ++j) {
      if (p[j] != 0) {
        row_out[b * NT + k] = (float)(j - 1);
        col_out[b * NT + k] = (float)(p[j] - 1);
        ++k;
      }
    }
  }
}

// ---------------------------------------------------------------------------
extern "C" void kernel_launch(void* const* d_in, const int* in_sizes, int n_in,
                              void* d_out, int out_size, void* d_ws,
                              size_t ws_size, hipStream_t stream) {
  const float* pred_logits = (const float*)d_in[0];
  const float* pred_boxes = (const float*)d_in[1];
  const float* pred_angles = (const float*)d_in[2];
  const int* tgt_ids = (const int*)d_in[3];
  const float* tgt_bbox = (const float*)d_in[4];
  const float* tgt_angles = (const float*)d_in[5];
  float* out = (float*)d_out;
  float* probs = (float*)d_ws;  // PTOT*NC floats = 1 MB scratch

  hm_softmax16<<<PTOT / 256, 256, 0, stream>>>(pred_logits, probs);

  const int tiles = (PTOT / 16) * (TTOT / 16);  // 131072 tiles, 8 waves/block
  hm_cost<<<tiles / 8, 256, 0, stream>>>(probs, pred_boxes, pred_angles,
                                         tgt_ids, tgt_bbox, tgt_angles, out);

  float* row_out = out + (size_t)PTOT * TTOT;
  float* col_out = row_out + BS * NT;
  size_t shmem = (size_t)(NT * NQ + 2 * (NQ + 1) + (NT + 1)) * sizeof(float) +
                 (size_t)3 * (NQ + 1) * sizeof(int);
  hm_lsa<<<BS, 32, shmem, stream>>>(out, row_out, col_out);
}